// LightGCN_5866925326460
// MI455X (gfx1250) — compile-verified
//
#include <hip/hip_runtime.h>
#include <hip/hip_bf16.h>

#define DIM 64

typedef __attribute__((ext_vector_type(2))) float v2f;
typedef __attribute__((ext_vector_type(8))) float v8f;

__device__ __forceinline__ float waveSum32(float v) {
#pragma unroll
  for (int off = 16; off >= 1; off >>= 1)
    v += __shfl_xor(v, off, 32);
  return v;
}

__device__ __forceinline__ void atomAddF(float* p, float v) {
  // hardware global_atomic_add_f32 (no return -> STOREcnt path)
  unsafeAtomicAdd(p, v);
}

// ---------------- row-wise L2 normalize (one wave per 64-wide row) ----------
__global__ void k_rownorm(const float* __restrict__ in, float* __restrict__ out,
                          int rows, float scale) {
  const int lane = threadIdx.x & 31;
  const int row = blockIdx.x * (blockDim.x >> 5) + (threadIdx.x >> 5);
  if (row >= rows) return;
  const float* p = in + (size_t)row * DIM;
  float a0 = p[lane] * scale;
  float a1 = p[lane + 32] * scale;
  float ss = waveSum32(a0 * a0 + a1 * a1);
  float inv = 1.0f / fmaxf(sqrtf(ss), 1e-12f);
  float* q = out + (size_t)row * DIM;
  q[lane] = a0 * inv;
  q[lane + 32] = a1 * inv;
}

// ---------------- item projection: l2norm(feat @ W^T + b) via f32 WMMA ------
// One wave per 16-item tile. feat[i] = [audio[i] (64) | artist+album (64)].
// D = A(16x4) x B(4x16) + C, K-loop over 128 in steps of 4, 4 N-tiles of 16.
__global__ void __launch_bounds__(128)
k_item_proj(const float* __restrict__ audio, const float* __restrict__ artist_w,
            const float* __restrict__ album_w, const float* __restrict__ projW,
            const float* __restrict__ projB, const int* __restrict__ artist_ids,
            const int* __restrict__ album_ids, float* __restrict__ x,
            int I, int U) {
  __shared__ float featS[4 * 16 * 128];  // per-wave 16x128 feat tile (8 KB)
  __shared__ float outS[4 * 16 * 64];    // per-wave 16x64 projected tile (4 KB)

  const int lane = threadIdx.x & 31;
  const int wave = threadIdx.x >> 5;
  const int tile = blockIdx.x * 4 + wave;
  const int nTiles = (I + 15) / 16;
  if (tile >= nTiles) return;  // no cross-wave barriers anywhere below

  float* fw = featS + wave * (16 * 128);
  float* ow = outS + wave * (16 * 64);
  const int base = tile * 16;

  // --- stage 16x128 feat tile into LDS (fused artist/album gather) ---
#pragma unroll 4
  for (int r = 0; r < 16; ++r) {
    int item = base + r;
    if (item >= I) item = I - 1;  // dup pad (I % 16 == 0 in practice)
    float4 v;
    if (lane < 16) {
      v = *(const float4*)(audio + (size_t)item * DIM + lane * 4);
    } else {
      const int c = (lane - 16) * 4;
      const int aid = artist_ids[item];
      const int bid = album_ids[item];
      float4 a = *(const float4*)(artist_w + (size_t)aid * DIM + c);
      float4 b = *(const float4*)(album_w + (size_t)bid * DIM + c);
      v = make_float4(a.x + b.x, a.y + b.y, a.z + b.z, a.w + b.w);
    }
    *(float4*)(fw + r * 128 + lane * 4) = v;
  }
  asm volatile("s_wait_dscnt 0" ::: "memory");  // intra-wave LDS WAR->RAW

  // --- K-loop of fp32 WMMA 16x16x4 ---
  v8f c0 = {}, c1 = {}, c2 = {}, c3 = {};
  const int m = lane & 15;               // A: M = lane % 16
  const int khalf = (lane >> 4) << 1;    // A/B: K offset 0 (lo half) / 2 (hi)
  const int n0 = lane & 15;              // B/D: N = lane % 16 (+16*nt)
  for (int kb = 0; kb < 128; kb += 4) {
    const int ko = kb + khalf;
    v2f a  = *(const v2f*)(fw + m * 128 + ko);
    v2f b0 = *(const v2f*)(projW + (size_t)(n0 +  0) * 128 + ko);
    v2f b1 = *(const v2f*)(projW + (size_t)(n0 + 16) * 128 + ko);
    v2f b2 = *(const v2f*)(projW + (size_t)(n0 + 32) * 128 + ko);
    v2f b3 = *(const v2f*)(projW + (size_t)(n0 + 48) * 128 + ko);
    c0 = __builtin_amdgcn_wmma_f32_16x16x4_f32(false, a, false, b0, (short)0, c0, false, false);
    c1 = __builtin_amdgcn_wmma_f32_16x16x4_f32(false, a, false, b1, (short)0, c1, false, false);
    c2 = __builtin_amdgcn_wmma_f32_16x16x4_f32(false, a, false, b2, (short)0, c2, false, false);
    c3 = __builtin_amdgcn_wmma_f32_16x16x4_f32(false, a, false, b3, (short)0, c3, false, false);
  }

  // --- bias add, stage 16x64 result in LDS ---
  const int mh = (lane >> 4) << 3;  // D: VGPR v holds M = v + 8*(lane>=16)
#pragma unroll
  for (int nt = 0; nt < 4; ++nt) {
    v8f cc = (nt == 0) ? c0 : (nt == 1) ? c1 : (nt == 2) ? c2 : c3;
    const float bias = projB[nt * 16 + n0];
#pragma unroll
    for (int v = 0; v < 8; ++v)
      ow[(v + mh) * 64 + nt * 16 + n0] = cc[v] + bias;
  }
  asm volatile("s_wait_dscnt 0" ::: "memory");

  // --- per-row L2 normalize, write item embeddings into x[U + item] ---
  for (int r = 0; r < 16; ++r) {
    const int item = base + r;
    if (item >= I) break;
    float a0 = ow[r * 64 + lane];
    float a1 = ow[r * 64 + lane + 32];
    float ss = waveSum32(a0 * a0 + a1 * a1);
    float inv = 1.0f / fmaxf(sqrtf(ss), 1e-12f);
    float* q = x + (size_t)(U + item) * DIM;
    q[lane] = a0 * inv;
    q[lane + 32] = a1 * inv;
  }
}

// ---------------- degree / dinv ---------------------------------------------
__global__ void k_init_deg(float* __restrict__ deg, int N) {
  int i = blockIdx.x * blockDim.x + threadIdx.x;
  if (i < N) deg[i] = 1.0f;  // self-loop weight folded in
}

__global__ void k_edge_deg(const float* __restrict__ ew, const int* __restrict__ ui,
                           const int* __restrict__ ii, float* __restrict__ deg,
                           int E, int U) {
  int i = blockIdx.x * blockDim.x + threadIdx.x;
  const int stride = gridDim.x * blockDim.x;
  for (; i < E; i += stride) {
    const float w = fmaxf(ew[i], 1e-6f);
    atomAddF(&deg[ui[i]], w);
    atomAddF(&deg[U + ii[i]], w);
  }
}

__global__ void k_dinv(float* __restrict__ deg, int N) {
  int i = blockIdx.x * blockDim.x + threadIdx.x;
  if (i < N) {
    const float d = deg[i];
    deg[i] = (d > 0.0f) ? rsqrtf(d) : 0.0f;  // in-place deg -> dinv
  }
}

// ---------------- propagation -----------------------------------------------
__global__ void k_layer_init(const float* __restrict__ x, const float* __restrict__ dinv,
                             float* __restrict__ xn, int N) {
  int i = blockIdx.x * blockDim.x + threadIdx.x;
  if (i < N * DIM) {
    const float dv = dinv[i >> 6];
    xn[i] = dv * dv * x[i];  // self-loop message: nrm = dinv[n]*1*dinv[n]
  }
}

// one wave per edge; lanes cover the 64-wide rows (2 floats/lane, coalesced)
__global__ void k_edge_prop(const float* __restrict__ x, float* __restrict__ xn,
                            const float* __restrict__ dinv, const float* __restrict__ ew,
                            const int* __restrict__ ui, const int* __restrict__ ii,
                            int E, int U) {
  const int lane = threadIdx.x & 31;
  int w = blockIdx.x * (blockDim.x >> 5) + (threadIdx.x >> 5);
  const int nw = gridDim.x * (blockDim.x >> 5);
  for (int e = w; e < E; e += nw) {
    const int u = ui[e];
    const int it = U + ii[e];
    const float wt = fmaxf(ew[e], 1e-6f);
    const float coef = dinv[u] * wt * dinv[it];
    const float* xu = x + (size_t)u * DIM;
    const float* xi = x + (size_t)it * DIM;
    const float u0 = xu[lane], u1 = xu[lane + 32];
    const float i0 = xi[lane], i1 = xi[lane + 32];
    float* yu = xn + (size_t)u * DIM;
    float* yi = xn + (size_t)it * DIM;
    atomAddF(yi + lane,      coef * u0);
    atomAddF(yi + lane + 32, coef * u1);
    atomAddF(yu + lane,      coef * i0);
    atomAddF(yu + lane + 32, coef * i1);
  }
}

// ---------------- elementwise helpers ---------------------------------------
__global__ void k_copy(const float* __restrict__ s, float* __restrict__ d, int n) {
  int i = blockIdx.x * blockDim.x + threadIdx.x;
  if (i < n) d[i] = s[i];
}
__global__ void k_accum(const float* __restrict__ s, float* __restrict__ d, int n) {
  int i = blockIdx.x * blockDim.x + threadIdx.x;
  if (i < n) d[i] += s[i];
}

// ---------------- launcher ---------------------------------------------------
extern "C" void kernel_launch(void* const* d_in, const int* in_sizes, int n_in,
                              void* d_out, int out_size, void* d_ws, size_t ws_size,
                              hipStream_t stream) {
  const float* user_w     = (const float*)d_in[0];
  const float* item_audio = (const float*)d_in[1];
  const float* artist_w   = (const float*)d_in[2];
  const float* album_w    = (const float*)d_in[3];
  const float* proj_W     = (const float*)d_in[4];
  const float* proj_b     = (const float*)d_in[5];
  const float* edge_w     = (const float*)d_in[6];
  const int*   u_idx      = (const int*)d_in[7];
  const int*   i_idx      = (const int*)d_in[8];
  const int*   artist_ids = (const int*)d_in[9];
  const int*   album_ids  = (const int*)d_in[10];
  float* out = (float*)d_out;  // doubles as the running accumulator `acc`

  const int U = in_sizes[0] / DIM;
  const int I = in_sizes[1] / DIM;
  const int E = in_sizes[6];
  const int N = U + I;

  float* ws  = (float*)d_ws;
  float* x   = ws;                        // [N, 64]
  float* xn  = ws + (size_t)N * DIM;      // [N, 64]
  float* deg = ws + (size_t)2 * N * DIM;  // [N]  (becomes dinv in place)

  // L0 embeddings
  k_rownorm<<<(U + 7) / 8, 256, 0, stream>>>(user_w, x, U, 1.0f);
  const int tiles = (I + 15) / 16;
  k_item_proj<<<(tiles + 3) / 4, 128, 0, stream>>>(
      item_audio, artist_w, album_w, proj_W, proj_b, artist_ids, album_ids,
      x, I, U);

  // gcn_norm (identical every layer): deg -> dinv
  k_init_deg<<<(N + 255) / 256, 256, 0, stream>>>(deg, N);
  k_edge_deg<<<4096, 256, 0, stream>>>(edge_w, u_idx, i_idx, deg, E, U);
  k_dinv<<<(N + 255) / 256, 256, 0, stream>>>(deg, N);

  // acc = x
  const int total = N * DIM;
  const int tb = (total + 255) / 256;
  k_copy<<<tb, 256, 0, stream>>>(x, out, total);

  // L = 3 propagation layers, ping-pong x <-> xn
  for (int l = 0; l < 3; ++l) {
    k_layer_init<<<tb, 256, 0, stream>>>(x, deg, xn, N);
    k_edge_prop<<<8192, 256, 0, stream>>>(x, xn, deg, edge_w, u_idx, i_idx, E, U);
    k_accum<<<tb, 256, 0, stream>>>(xn, out, total);
    float* t = x; x = xn; xn = t;
  }

  // out = l2norm(acc / 4), in place (each wave owns its row)
  k_rownorm<<<(N + 7) / 8, 256, 0, stream>>>(out, out, N, 0.25f);
}